// GCN_46952582480547
// MI455X (gfx1250) — compile-verified
//
#include <hip/hip_runtime.h>

typedef __attribute__((ext_vector_type(2))) float v2f;
typedef __attribute__((ext_vector_type(8))) float v8f;

#define N_FEAT 128

// ---------------- utility kernels ----------------

__global__ void zero_f32(float* __restrict__ p, int n) {
  int i = blockIdx.x * blockDim.x + threadIdx.x;
  if (i < n) p[i] = 0.0f;
}

// Integer degree histograms (deterministic, exact).
__global__ void degree_count(const int* __restrict__ src, const int* __restrict__ dst,
                             unsigned int* __restrict__ cnt_out,
                             unsigned int* __restrict__ cnt_in, int nE) {
  int e = blockIdx.x * blockDim.x + threadIdx.x;
  if (e < nE) {
    atomicAdd(&cnt_out[src[e]], 1u);
    atomicAdd(&cnt_in[dst[e]], 1u);
  }
}

// In-place: uint count -> rsqrt(max(count,1)) as float.
__global__ void rsqrt_deg(float* __restrict__ buf, int n) {
  int i = blockIdx.x * blockDim.x + threadIdx.x;
  if (i < n) {
    unsigned int c = ((const unsigned int*)buf)[i];
    float d = (float)(c == 0u ? 1u : c);
    buf[i] = rsqrtf(d);
  }
}

// out[i] = x[i] * s[i / 128]   (row scaling by rsqrt(deg_out))
__global__ void scale_rows(const float* __restrict__ x, const float* __restrict__ s,
                           float* __restrict__ out, int total) {
  int i = blockIdx.x * blockDim.x + threadIdx.x;
  if (i < total) out[i] = x[i] * s[i >> 7];
}

// agg[dst[e]][f] += h[src[e]][f]  (one thread per (edge, feature))
__global__ void scatter_feat128(const float* __restrict__ h, const int* __restrict__ src,
                                const int* __restrict__ dst, float* __restrict__ agg,
                                int nE) {
  long long t = (long long)blockIdx.x * blockDim.x + threadIdx.x;
  if (t < (long long)nE * N_FEAT) {
    int e = (int)(t >> 7);
    int f = (int)(t & 127);
    unsafeAtomicAdd(&agg[(long long)dst[e] * N_FEAT + f],
                    h[(long long)src[e] * N_FEAT + f]);
  }
}

// ---------------- WMMA fp32 GEMM: out = relu((A .* rsd_in) @ W + bias) ----------------
// A: nNodes x 128 row-major, W: 128 x 128 row-major, out: nNodes x 128.
// One block per 16-row M tile; 8 waves, each wave owns one 16-col N tile.
// V_WMMA_F32_16X16X4_F32 layouts (wave32):
//   A 16x4 : lane L -> M = L%16; VGPR0 holds K = (L<16 ? 0 : 2), VGPR1 K = (L<16 ? 1 : 3)
//   B 4x16 : lane L -> N = L%16; VGPR0 holds K = (L<16 ? 0 : 2), VGPR1 K = (L<16 ? 1 : 3)
//   C 16x16: lane L, VGPR v -> M = v + (L<16 ? 0 : 8), N = L%16
__global__ void gemm128_relu_wmma(const float* __restrict__ A,
                                  const float* __restrict__ rsdin,
                                  const float* __restrict__ W,
                                  const float* __restrict__ bias,
                                  float* __restrict__ out, int nNodes) {
  __shared__ float As[16][132];  // 132-stride: 16 rows hit banks 0,4,...,60 -> conflict-free

  const int m0 = blockIdx.x * 16;

  // Cooperative, coalesced A-tile load fused with rsqrt(deg_in) scaling.
  for (int idx = threadIdx.x; idx < 16 * N_FEAT; idx += blockDim.x) {
    int r = idx >> 7, c = idx & 127;
    int row = m0 + r;
    As[r][c] = (row < nNodes) ? A[(long long)row * N_FEAT + c] * rsdin[row] : 0.0f;
  }
  __syncthreads();

  const int lane = threadIdx.x & 31;
  const int wv   = threadIdx.x >> 5;   // N tile id 0..7
  const int hi   = lane >> 4;          // upper half-wave?
  const int l16  = lane & 15;
  const int n    = wv * 16 + l16;

  v8f acc = {0.f, 0.f, 0.f, 0.f, 0.f, 0.f, 0.f, 0.f};

  #pragma unroll
  for (int kb = 0; kb < N_FEAT; kb += 4) {
    const int k0 = kb + 2 * hi;                 // K of VGPR0; VGPR1 is k0+1
    v2f a = *(const v2f*)&As[l16][k0];          // 8B-aligned LDS read (ds_load b64)
    v2f b;
    b.x = W[(long long)k0 * N_FEAT + n];
    b.y = W[(long long)(k0 + 1) * N_FEAT + n];
    acc = __builtin_amdgcn_wmma_f32_16x16x4_f32(
        /*neg_a=*/false, a, /*neg_b=*/false, b,
        /*c_mod=*/(short)0, acc, /*reuse_a=*/false, /*reuse_b=*/false);
  }

  const float bn = bias[n];
  const int mbase = m0 + (hi ? 8 : 0);
  #pragma unroll
  for (int v = 0; v < 8; ++v) {
    const int m = mbase + v;
    if (m < nNodes) {
      const float val = acc[v] + bn;
      out[(long long)m * N_FEAT + n] = val > 0.f ? val : 0.f;
    }
  }
}

// ---------------- layer 2: h2[i][c] = rsd_out[i] * sum_k h[i][k] * W2[k][c] ----------------
__global__ void proj2(const float* __restrict__ h, const float* __restrict__ rsdout,
                      const float* __restrict__ W2, float* __restrict__ h2, int nNodes) {
  const int wv = threadIdx.x >> 5;
  const int lane = threadIdx.x & 31;
  const int node = blockIdx.x * (blockDim.x >> 5) + wv;
  if (node >= nNodes) return;
  float s0 = 0.f, s1 = 0.f;
  const float* row = h + (long long)node * N_FEAT;
  #pragma unroll
  for (int k = lane; k < N_FEAT; k += 32) {
    float v = row[k];
    s0 += v * W2[2 * k];
    s1 += v * W2[2 * k + 1];
  }
  #pragma unroll
  for (int off = 16; off > 0; off >>= 1) {
    s0 += __shfl_down(s0, off, 32);
    s1 += __shfl_down(s1, off, 32);
  }
  if (lane == 0) {
    float sc = rsdout[node];
    h2[2 * node]     = s0 * sc;
    h2[2 * node + 1] = s1 * sc;
  }
}

__global__ void scatter_feat2(const float* __restrict__ h2, const int* __restrict__ src,
                              const int* __restrict__ dst, float* __restrict__ agg2,
                              int nE) {
  int t = blockIdx.x * blockDim.x + threadIdx.x;
  if (t < nE * 2) {
    int e = t >> 1, c = t & 1;
    unsafeAtomicAdd(&agg2[2 * dst[e] + c], h2[2 * src[e] + c]);
  }
}

__global__ void finalize(const float* __restrict__ agg2, const float* __restrict__ rsdin,
                         const float* __restrict__ b2, float* __restrict__ out, int n2) {
  int i = blockIdx.x * blockDim.x + threadIdx.x;
  if (i < n2) out[i] = agg2[i] * rsdin[i >> 1] + b2[i & 1];
}

// ---------------- launch ----------------

extern "C" void kernel_launch(void* const* d_in, const int* in_sizes, int n_in,
                              void* d_out, int out_size, void* d_ws, size_t ws_size,
                              hipStream_t stream) {
  const float* x   = (const float*)d_in[0];
  const int*   src = (const int*)d_in[1];
  const int*   dst = (const int*)d_in[2];
  const float* W1  = (const float*)d_in[3];
  const float* b1  = (const float*)d_in[4];
  const float* W2  = (const float*)d_in[5];
  const float* b2  = (const float*)d_in[6];

  const int nNodes = in_sizes[0] / N_FEAT;
  const int nE     = in_sizes[1];

  // Workspace layout (floats): [rsd_out n][rsd_in n][agg1 128n][agg2 2n][hs 128n][h2 2n]
  float* rsd_out = (float*)d_ws;
  float* rsd_in  = rsd_out + nNodes;
  float* agg1    = rsd_in + nNodes;
  float* agg2    = agg1 + (size_t)nNodes * N_FEAT;
  float* hs      = agg2 + (size_t)nNodes * 2;
  float* h2      = hs + (size_t)nNodes * N_FEAT;

  const int B = 256;

  // 1) zero counts + accumulators (contiguous 132n floats)
  int zn = nNodes * (2 + N_FEAT + 2);
  zero_f32<<<(zn + B - 1) / B, B, 0, stream>>>(rsd_out, zn);

  // 2) degrees -> rsqrt
  degree_count<<<(nE + B - 1) / B, B, 0, stream>>>(
      src, dst, (unsigned int*)rsd_out, (unsigned int*)rsd_in, nE);
  rsqrt_deg<<<(2 * nNodes + B - 1) / B, B, 0, stream>>>(rsd_out, 2 * nNodes);

  // 3) layer 1: scale, scatter, GEMM+bias+relu
  int tx = nNodes * N_FEAT;
  scale_rows<<<(tx + B - 1) / B, B, 0, stream>>>(x, rsd_out, hs, tx);
  long long te = (long long)nE * N_FEAT;
  scatter_feat128<<<(int)((te + B - 1) / B), B, 0, stream>>>(hs, src, dst, agg1, nE);
  gemm128_relu_wmma<<<(nNodes + 15) / 16, B, 0, stream>>>(agg1, rsd_in, W1, b1, hs, nNodes);

  // 4) layer 2: project to 2 classes, scatter, finalize
  proj2<<<(nNodes + 7) / 8, B, 0, stream>>>(hs, rsd_out, W2, h2, nNodes);
  scatter_feat2<<<(nE * 2 + B - 1) / B, B, 0, stream>>>(h2, src, dst, agg2, nE);
  finalize<<<(2 * nNodes + B - 1) / B, B, 0, stream>>>(agg2, rsd_in, b2, (float*)d_out,
                                                      2 * nNodes);
}